// SimpleRNN_30520037606069
// MI455X (gfx1250) — compile-verified
//
#include <hip/hip_runtime.h>

// SimpleRNN on gfx1250 (MI455X), fp32 end-to-end via V_WMMA_F32_16X16X4_F32.
//
// One wave32 per 16-row batch tile (256 tiles). Per timestep:
//   two independent WMMA accumulator chains (depth 3 instead of 5):
//     accA = x_t W_ih^T + bias  -> + h chunks K=0..3, K=4..7
//     accB =                       h chunks K=8..11, K=12..15 (C = 0)
//   h = v_tanh_f32(accA + accB)  (hardware TRANS op, branch-free fallback)
//   D-layout -> LDS -> A-layout round trip for the next step's A operand
//   output head (O=2) per-lane 16-tap dot from the LDS tile, 4-way partials.
//
// Layout assumptions (CDNA5 ISA 7.12.2):
//   A 16x4 f32:  lanes 0-15: M=lane, V0=K0, V1=K1 ; lanes 16-31: M=lane-16, V0=K2, V1=K3
//   B 4x16 f32 (mirror of A): lanes 0-15: N=lane, V0=K0, V1=K1 ; lanes 16-31: N=lane-16, V0=K2, V1=K3
//   C/D 16x16 f32: lane l, VGPR r -> [M = r + 8*(l/16)][N = l%16]

typedef __attribute__((ext_vector_type(2))) float v2f;
typedef __attribute__((ext_vector_type(8))) float v8f;

#define B_SZ 4096
#define T_SZ 2048
#define I_SZ 3
#define H_SZ 16
#define O_SZ 2
#define LDS_STRIDE 20   // padded so lane-halves hit disjoint bank sets

// Branch-free tanh: prefer the gfx1250 hardware V_TANH_F32 (single TRANS op).
__device__ __forceinline__ float fast_tanh(float v) {
#if defined(__has_builtin) && __has_builtin(__builtin_amdgcn_tanhf)
    return __builtin_amdgcn_tanhf(v);
#else
    const float a = __builtin_fabsf(v);
    const float e = __builtin_amdgcn_exp2f(a * 2.88539008177792681472f); // 2/ln(2)
    const float r = __builtin_amdgcn_rcpf(e + 1.0f);
    const float t = __builtin_fmaf(-2.0f, r, 1.0f);
    return __builtin_copysignf(t, v);
#endif
}

__global__ __launch_bounds__(32)
void rnn_wmma_f32_kernel(const float* __restrict__ x,
                         const float* __restrict__ W_ih,
                         const float* __restrict__ W_hh,
                         const float* __restrict__ b_ih,
                         const float* __restrict__ b_hh,
                         const float* __restrict__ W_out,
                         const float* __restrict__ b_out,
                         float* __restrict__ out)
{
    __shared__ float hbuf[16 * LDS_STRIDE];

    const int lane = threadIdx.x;       // 0..31
    const int lo   = lane & 15;         // row/col within tile
    const int hi   = lane >> 4;         // which K-pair half / which output o

    const int rowbase = blockIdx.x * 16;
    const int row     = rowbase + lo;

    // ---- loop-invariant operand fragments -------------------------------
    // B fragment of W_ih^T (4x16, B[k][n] = W_ih[n][k], row k=3 is zero):
    v2f Bih;
    Bih.x = W_ih[lo * I_SZ + 2 * hi];               // hi=0: W_ih[n][0], hi=1: W_ih[n][2]
    Bih.y = hi ? 0.0f : W_ih[lo * I_SZ + 1];        // hi=0: W_ih[n][1], hi=1: zero row K=3

    // B fragments of W_hh^T (16x16 as four 4x16 chunks; B[k][n] = W_hh[n][k]):
    v2f Bhh[4];
    #pragma unroll
    for (int c = 0; c < 4; ++c) {
        const int k = 4 * c + 2 * hi;
        Bhh[c].x = W_hh[lo * H_SZ + k];
        Bhh[c].y = W_hh[lo * H_SZ + k + 1];
    }

    // Bias tile C[m][n] = b_ih[n] + b_hh[n]: every VGPR holds column n = lo.
    const float bias = b_ih[lo] + b_hh[lo];
    v8f Cb, Cz;
    #pragma unroll
    for (int r = 0; r < 8; ++r) { Cb[r] = bias; Cz[r] = 0.0f; }

    // Output-head row for this lane's o = hi.
    float wout[16];
    #pragma unroll
    for (int n = 0; n < 16; ++n) wout[n] = W_out[hi * H_SZ + n];
    const float bo = b_out[hi];

    // ---- h_0 = 0 --------------------------------------------------------
    #pragma unroll
    for (int r = 0; r < 8; ++r) hbuf[(r + 8 * hi) * LDS_STRIDE + lo] = 0.0f;
    __syncthreads();

    const float* xp   = x   + (size_t)row * T_SZ * I_SZ;
    float*       outp = out + (size_t)row * T_SZ * O_SZ;

    v8f h = Cz;  // final-state carrier (always overwritten, T > 0)

    for (int t = 0; t < T_SZ; ++t) {
        // A fragment of x_t (16x4): lanes 0-15 hold (K0,K1), 16-31 hold (K2, 0).
        const size_t xo = (size_t)t * I_SZ;
        const float a0 = xp[xo + 2 * hi];
        const float a1 = xp[xo + 1];                 // always in-bounds
        v2f Ax; Ax.x = a0; Ax.y = hi ? 0.0f : a1;
        __builtin_prefetch(&xp[xo + 8 * I_SZ], 0, 3); // near-cache prefetch, ~8 steps ahead

        // A fragments of h_{t-1} from LDS (ds_load_2addr_b64 pairs, conflict-free).
        v2f Ah[4];
        #pragma unroll
        for (int c = 0; c < 4; ++c)
            Ah[c] = *(const v2f*)&hbuf[lo * LDS_STRIDE + 4 * c + 2 * hi];

        // Two independent WMMA chains (depth 3), summed afterwards:
        //   accA = x_t W_ih^T + bias + h[:,0:8]  W_hh^T[0:8,:]
        //   accB =                     h[:,8:16] W_hh^T[8:16,:]
        v8f accA = __builtin_amdgcn_wmma_f32_16x16x4_f32(
            false, Ax, false, Bih, (short)0, Cb, false, false);
        accA = __builtin_amdgcn_wmma_f32_16x16x4_f32(
            false, Ah[0], false, Bhh[0], (short)0, accA, false, false);
        accA = __builtin_amdgcn_wmma_f32_16x16x4_f32(
            false, Ah[1], false, Bhh[1], (short)0, accA, false, false);

        v8f accB = __builtin_amdgcn_wmma_f32_16x16x4_f32(
            false, Ah[2], false, Bhh[2], (short)0, Cz, false, false);
        accB = __builtin_amdgcn_wmma_f32_16x16x4_f32(
            false, Ah[3], false, Bhh[3], (short)0, accB, false, false);

        #pragma unroll
        for (int r = 0; r < 8; ++r) h[r] = fast_tanh(accA[r] + accB[r]);

        __syncthreads();   // every lane done reading old state (WAR)
        #pragma unroll
        for (int r = 0; r < 8; ++r) hbuf[(r + 8 * hi) * LDS_STRIDE + lo] = h[r];
        __syncthreads();   // new state visible to all lanes (RAW)

        // Output head: lane (lo, hi) -> out[row][t][hi]; 4-way partial sums.
        float s0 = bo, s1 = 0.0f, s2 = 0.0f, s3 = 0.0f;
        #pragma unroll
        for (int n = 0; n < 4; ++n) {
            s0 = __builtin_fmaf(hbuf[lo * LDS_STRIDE + n     ], wout[n     ], s0);
            s1 = __builtin_fmaf(hbuf[lo * LDS_STRIDE + n + 4 ], wout[n + 4 ], s1);
            s2 = __builtin_fmaf(hbuf[lo * LDS_STRIDE + n + 8 ], wout[n + 8 ], s2);
            s3 = __builtin_fmaf(hbuf[lo * LDS_STRIDE + n + 12], wout[n + 12], s3);
        }
        outp[(size_t)t * O_SZ + hi] = fast_tanh((s0 + s1) + (s2 + s3));
    }

    // Final hidden state h_T -> d_out tail [B, H] (D layout scatter).
    float* hT = out + (size_t)B_SZ * T_SZ * O_SZ;
    #pragma unroll
    for (int r = 0; r < 8; ++r)
        hT[(size_t)(rowbase + r + 8 * hi) * H_SZ + lo] = h[r];
}

extern "C" void kernel_launch(void* const* d_in, const int* in_sizes, int n_in,
                              void* d_out, int out_size, void* d_ws, size_t ws_size,
                              hipStream_t stream) {
    (void)in_sizes; (void)n_in; (void)d_ws; (void)ws_size; (void)out_size;
    const float* x     = (const float*)d_in[0];
    const float* W_ih  = (const float*)d_in[1];
    const float* W_hh  = (const float*)d_in[2];
    const float* b_ih  = (const float*)d_in[3];
    const float* b_hh  = (const float*)d_in[4];
    const float* W_out = (const float*)d_in[5];
    const float* b_out = (const float*)d_in[6];
    float* out = (float*)d_out;

    dim3 grid(B_SZ / 16);   // 256 batch tiles, one wave32 each
    dim3 block(32);
    rnn_wmma_f32_kernel<<<grid, block, 0, stream>>>(
        x, W_ih, W_hh, b_ih, b_hh, W_out, b_out, out);
}